// TokenMixer_57939108823099
// MI455X (gfx1250) — compile-verified
//
#include <hip/hip_runtime.h>
#include <hip/hip_bf16.h>

// ---------------------------------------------------------------------------
// TokenMixer for MI455X (gfx1250): pure bandwidth problem (~226MB HBM traffic
// => ~10us roofline at 23.3 TB/s). No matmul in the op; use b128 vector memory,
// non-temporal stores for the write-once 170MB output, async global<->LDS DMA
// copies (CDNA5 ASYNCcnt path) for the dominant pure-copy lines, and
// global_prefetch for the gathered mix sources.
// ---------------------------------------------------------------------------

#define NUM_CLASSES 174
#define KMIX 3
#define BB 16
#define TT 16
#define NR_BOX 8
#define CELLS 9
#define DD 768

#define LINE_F   (CELLS * DD)        // 6912 floats per (sample,t,box) line
#define LINE_F4  (LINE_F / 4)        // 1728 float4 per line
#define MIX_OBJ_ELEMS (BB * KMIX * TT * NR_BOX * LINE_F)  // 42,467,328 floats

typedef __attribute__((ext_vector_type(4))) float f32x4;
typedef __attribute__((ext_vector_type(4))) int   i32x4;

#ifndef __has_builtin
#define __has_builtin(x) 0
#endif

#if __has_builtin(__builtin_amdgcn_global_load_async_to_lds_b128) && \
    __has_builtin(__builtin_amdgcn_global_store_async_from_lds_b128) && \
    __has_builtin(__builtin_amdgcn_s_wait_asynccnt)
#define HAVE_ASYNC_LDS 1
#else
#define HAVE_ASYNC_LDS 0
#endif

#if HAVE_ASYNC_LDS
// Builtin signature (from hipcc diagnostics): args are pointers to 16B int
// vectors in addrspace(1) (global) and addrspace(3) (LDS).
typedef __attribute__((address_space(1))) i32x4 gas_i32x4;
typedef __attribute__((address_space(3))) i32x4 las_i32x4;

// CPol immediate: TH in bits [2:0] (store TH: 0=RT, 1=NT), scope in [4:3].
#define CPOL_RT 0
#define CPOL_NT 1

// generic->AS1: same 64-bit value.  generic->AS3: low 32 bits are the LDS offset.
__device__ __forceinline__ gas_i32x4* as_global(const void* p) {
  return (gas_i32x4*)(unsigned long long)p;
}
__device__ __forceinline__ las_i32x4* as_local(const void* p) {
  return (las_i32x4*)(unsigned int)(unsigned long long)p;
}
#endif

// ---------------------------------------------------------------------------
// Prep: per-(i,k,n) blend coefficients (384 float4 into d_ws) + mix_label
// (B*K*NUM_CLASSES floats at the tail of d_out).  Cumulative mixing recurrence:
//   for j in 0..k: if slots[i][j]==n: v = lam*v + (1-lam)*sel_j
// => a *= lam; all existing b_* *= lam; b_j += (1-lam).
// ---------------------------------------------------------------------------
__global__ __launch_bounds__(256) void tokenmixer_prep_kernel(
    const int* __restrict__ labels, const float* __restrict__ lam,
    const int* __restrict__ select_batch, const int* __restrict__ slots,
    f32x4* __restrict__ coeffs, float* __restrict__ mix_label) {
  const int ik = blockIdx.x;          // i*KMIX + k
  const int i = ik / KMIX;
  const int k = ik % KMIX;
  const float lam_i = lam[i];
  const int tid = threadIdx.x;

  if (tid < NR_BOX) {
    const int n = tid;
    float a = 1.0f, b0 = 0.0f, b1 = 0.0f, b2 = 0.0f;
    for (int j = 0; j <= k; ++j) {
      if (slots[i * KMIX + j] == n) {
        const float add = 1.0f - lam_i;
        a *= lam_i; b0 *= lam_i; b1 *= lam_i; b2 *= lam_i;
        if (j == 0) b0 += add; else if (j == 1) b1 += add; else b2 += add;
      }
    }
    f32x4 c = {a, b0, b1, b2};
    coeffs[ik * NR_BOX + n] = c;
  }

  // labs[i,k,:] = onehot(labels[i])*lam + onehot(labels[sb[i,k]])*(1-lam)
  const int own = labels[i];
  const int other = labels[select_batch[i * KMIX + k]];
  for (int c = tid; c < NUM_CLASSES; c += blockDim.x) {
    float v = (c == own ? lam_i : 0.0f) + (c == other ? (1.0f - lam_i) : 0.0f);
    mix_label[ik * NUM_CLASSES + c] = v;
  }
}

// ---------------------------------------------------------------------------
// Main mix: one block per output line (i,k,t,n) => 6144 blocks x 256 threads,
// each line is 1728 float4.  All control flow is uniform per block.
//   pure copy (a==1, b==0):  async DMA global->LDS->global (ASYNCcnt path),
//                            RT loads (line reused across k), NT stores.
//   blended:                 v = a*orig + sum_j b_j*sel_j, NT store.
// ---------------------------------------------------------------------------
__global__ __launch_bounds__(256) void tokenmixer_mix_kernel(
    const float* __restrict__ region, const int* __restrict__ select_batch,
    const int* __restrict__ select_box, const f32x4* __restrict__ coeffs,
    float* __restrict__ out) {
#if HAVE_ASYNC_LDS
  __shared__ f32x4 lbuf[LINE_F4];     // 27,648 B staging for DMA-style copy
#endif
  const int blk = blockIdx.x;                    // ((i*K+k)*T + t)*NR_BOX + n
  const int n = blk % NR_BOX;
  const int t = (blk / NR_BOX) % TT;
  const int ik = blk / (NR_BOX * TT);            // i*KMIX + k
  const int i = ik / KMIX;
  const int tid = threadIdx.x;

  const f32x4 cf = coeffs[ik * NR_BOX + n];
  const f32x4* __restrict__ orig =
      (const f32x4*)region + (size_t)((i * TT + t) * NR_BOX + n) * LINE_F4;
  f32x4* __restrict__ o = (f32x4*)out + (size_t)blk * LINE_F4;

  const bool purecopy =
      (cf.x == 1.0f) && (cf.y == 0.0f) && (cf.z == 0.0f) && (cf.w == 0.0f);

  if (purecopy) {
#if HAVE_ASYNC_LDS
    // DMA-style: async global->LDS, per-wave wait, async LDS->global.
    // Each lane's LDS slots are private to its wave, so no barrier needed.
    for (int idx = tid; idx < LINE_F4; idx += 256) {
      __builtin_amdgcn_global_load_async_to_lds_b128(
          as_global(orig + idx), as_local(&lbuf[idx]), 0, CPOL_RT);
    }
    __builtin_amdgcn_s_wait_asynccnt(0);
    for (int idx = tid; idx < LINE_F4; idx += 256) {
      __builtin_amdgcn_global_store_async_from_lds_b128(
          as_global(o + idx), as_local(&lbuf[idx]), 0, CPOL_NT);
    }
    __builtin_amdgcn_s_wait_asynccnt(0);
#else
    for (int idx = tid; idx < LINE_F4; idx += 256) {
      __builtin_nontemporal_store(orig[idx], &o[idx]);
    }
#endif
    return;
  }

  // Blended line: gather the 3 candidate source lines (only nonzero-coeff ones
  // are actually read).
  const f32x4* __restrict__ s0 =
      (const f32x4*)region +
      (size_t)((select_batch[i * KMIX + 0] * TT + t) * NR_BOX +
               select_box[i * KMIX + 0]) * LINE_F4;
  const f32x4* __restrict__ s1 =
      (const f32x4*)region +
      (size_t)((select_batch[i * KMIX + 1] * TT + t) * NR_BOX +
               select_box[i * KMIX + 1]) * LINE_F4;
  const f32x4* __restrict__ s2 =
      (const f32x4*)region +
      (size_t)((select_batch[i * KMIX + 2] * TT + t) * NR_BOX +
               select_box[i * KMIX + 2]) * LINE_F4;

  const bool u0 = (cf.y != 0.0f), u1 = (cf.z != 0.0f), u2 = (cf.w != 0.0f);
  if (u0) __builtin_prefetch(s0 + tid, 0, 0);   // global_prefetch_b8
  if (u1) __builtin_prefetch(s1 + tid, 0, 0);
  if (u2) __builtin_prefetch(s2 + tid, 0, 0);

  for (int idx = tid; idx < LINE_F4; idx += 256) {
    f32x4 v = orig[idx] * cf.x;
    if (u0) v += s0[idx] * cf.y;
    if (u1) v += s1[idx] * cf.z;
    if (u2) v += s2[idx] * cf.w;
    __builtin_nontemporal_store(v, &o[idx]);
  }
}

extern "C" void kernel_launch(void* const* d_in, const int* in_sizes, int n_in,
                              void* d_out, int out_size, void* d_ws, size_t ws_size,
                              hipStream_t stream) {
  // setup_inputs() order:
  //   0 region_vis_feas f32 [B,T,NR_BOX,CELLS,D]
  //   1 obj_indicator   f32 (unused by the reference outputs)
  //   2 labels          i32 [B]
  //   3 lam             f32 [B]
  //   4 select_batch    i32 [B,K]
  //   5 select_box      i32 [B,K]
  //   6 slots           i32 [B,K]
  const float* region = (const float*)d_in[0];
  const int* labels = (const int*)d_in[2];
  const float* lam = (const float*)d_in[3];
  const int* select_batch = (const int*)d_in[4];
  const int* select_box = (const int*)d_in[5];
  const int* slots = (const int*)d_in[6];

  float* out = (float*)d_out;                  // mix_obj, then mix_label
  f32x4* coeffs = (f32x4*)d_ws;                // 384 * 16B = 6 KB scratch

  (void)in_sizes; (void)n_in; (void)out_size; (void)ws_size;

  tokenmixer_prep_kernel<<<BB * KMIX, 256, 0, stream>>>(
      labels, lam, select_batch, slots, coeffs, out + MIX_OBJ_ELEMS);

  tokenmixer_mix_kernel<<<BB * KMIX * TT * NR_BOX, 256, 0, stream>>>(
      region, select_batch, select_box, coeffs, out);
}